// MaskedDecayAggregator_89945205113616
// MI455X (gfx1250) — compile-verified
//
#include <hip/hip_runtime.h>
#include <math.h>

typedef __attribute__((ext_vector_type(2))) float v2f;
typedef __attribute__((ext_vector_type(8))) float v8f;

#define TT 128
#define DD 768
#define DECAY_ALPHA 0.1f
#define EPS_W 1e-8f
#define LN_EPS 1e-5f

__device__ __forceinline__ float sanitize(float x) {
  // nan_to_num: NaN/+Inf/-Inf -> 0 (comparison is false for NaN)
  return (__builtin_fabsf(x) <= 3.402823e38f) ? x : 0.0f;
}

__global__ __launch_bounds__(256) void mda_kernel(
    const float* __restrict__ H, const int* __restrict__ valid_lens,
    const float* __restrict__ gamma, const float* __restrict__ beta,
    float* __restrict__ out)
{
  __shared__ float s_norm2[TT];
  __shared__ float s_w[TT];
  __shared__ float s_E[DD];
  __shared__ float s_red[8];
  __shared__ float s_bcast[3]; // [0]=1/max(wsum,eps), [1]=has_any, [2]=mean/var bcast

  const int bf   = blockIdx.x;
  const int tid  = threadIdx.x;
  const int lane = tid & 31;
  const int wv   = tid >> 5;
  const float* __restrict__ Hbf = H + (size_t)bf * (TT * DD);

  int L = valid_lens[bf];
  L = (L < 0) ? 0 : (L > TT ? TT : L);

  // ---------- Pass A: per-timestep squared norms (invalid -> +inf) ----------
  for (int t = wv; t < TT; t += 8) {
    float s = 0.0f;
    if (t < L) {
      const float* row = Hbf + t * DD;
      if (t + 8 < L) __builtin_prefetch(row + 8 * DD + lane * 4, 0, 0);
      #pragma unroll
      for (int c = 0; c < 6; ++c) {
        const float4 v = *(const float4*)(row + c * 128 + lane * 4);
        const float x = sanitize(v.x), y = sanitize(v.y);
        const float z = sanitize(v.z), w = sanitize(v.w);
        s += x * x + y * y + z * z + w * w;
      }
    }
    #pragma unroll
    for (int off = 16; off > 0; off >>= 1) s += __shfl_xor(s, off, 32);
    if (lane == 0) s_norm2[t] = (t < L) ? s : __builtin_inff();
  }
  __syncthreads();

  // ---------- Ranks (stable argsort-of-argsort) and decay weights ----------
  if (tid < TT) {
    const float my = s_norm2[tid];
    int rank = 0;
    #pragma unroll 4
    for (int j = 0; j < TT; ++j) {
      const float vj = s_norm2[j];
      rank += (vj < my || (vj == my && j < tid)) ? 1 : 0;
    }
    const int k_low = (int)(0.1f * (float)TT); // 12
    const int lo = (k_low < (L >> 1)) ? k_low : (L >> 1);
    const int hr = L - lo;
    const int hi = (hr > lo) ? hr : lo;
    const bool keep = (rank >= lo) && (rank < hi) && (tid < L);
    s_w[tid] = keep ? __expf(-DECAY_ALPHA * (float)(TT - 1 - tid)) : 0.0f;
  }
  __syncthreads();

  if (wv == 0) {
    float s = s_w[lane] + s_w[lane + 32] + s_w[lane + 64] + s_w[lane + 96];
    #pragma unroll
    for (int off = 16; off > 0; off >>= 1) s += __shfl_xor(s, off, 32);
    if (lane == 0) {
      s_bcast[0] = 1.0f / fmaxf(s, EPS_W);
      s_bcast[1] = (s > EPS_W) ? 1.0f : 0.0f;
    }
  }
  __syncthreads();

  // ---------- Pass B: E = Hm^T * w via V_WMMA_F32_16X16X4_F32 ----------
  // A (16x4 f32): lanes 0-15 hold M=lane {v0:K=0, v1:K=1}; lanes 16-31 {v0:K=2, v1:K=3}
  // B (4x16 f32): lanes 0-15 hold N=lane {v0:K=0, v1:K=1}; lanes 16-31 {v0:K=2, v1:K=3}
  // C (16x16):    VGPR r: lanes 0-15 -> M=r, lanes 16-31 -> M=8+r (all N columns equal here)
  const int half = lane >> 4;
  const int l16  = lane & 15;
  const float inv = s_bcast[0];

  for (int c = 0; c < 6; c += 2) {       // each wave owns a 96-wide d-stripe
    const int d0 = wv * 96 + c * 16;
    v8f acc0 = {0.f, 0.f, 0.f, 0.f, 0.f, 0.f, 0.f, 0.f};
    v8f acc1 = acc0;
    for (int t0 = 0; t0 < TT; t0 += 4) {
      const float w0 = s_w[t0], w1 = s_w[t0 + 1], w2 = s_w[t0 + 2], w3 = s_w[t0 + 3];
      // block-uniform skip: trimmed / invalid timestep groups contribute nothing
      if ((w0 == 0.0f) & (w1 == 0.0f) & (w2 == 0.0f) & (w3 == 0.0f)) continue;
      v2f b; b.x = half ? w2 : w0; b.y = half ? w3 : w1;
      const float* pa = Hbf + (size_t)(t0 + half * 2) * DD + d0 + l16;
      v2f a0, a1;
      a0.x = sanitize(pa[0]);   a0.y = sanitize(pa[DD]);
      a1.x = sanitize(pa[16]);  a1.y = sanitize(pa[DD + 16]);
      acc0 = __builtin_amdgcn_wmma_f32_16x16x4_f32(false, a0, false, b, (short)0,
                                                   acc0, false, false);
      acc1 = __builtin_amdgcn_wmma_f32_16x16x4_f32(false, a1, false, b, (short)0,
                                                   acc1, false, false);
    }
    if (l16 == 0) {   // lanes 0 and 16 hold N=0 column = the reduction result
      const int mb = half * 8;
      #pragma unroll
      for (int r = 0; r < 8; ++r) {
        s_E[d0 + mb + r]      = acc0[r] * inv;
        s_E[d0 + 16 + mb + r] = acc1[r] * inv;
      }
    }
  }
  __syncthreads();

  // ---------- LayerNorm over D, gated on has_any ----------
  const float e0 = s_E[tid], e1 = s_E[tid + 256], e2 = s_E[tid + 512];
  float s = e0 + e1 + e2;
  #pragma unroll
  for (int off = 16; off > 0; off >>= 1) s += __shfl_xor(s, off, 32);
  if (lane == 0) s_red[wv] = s;
  __syncthreads();
  if (tid == 0) {
    float tot = 0.f;
    #pragma unroll
    for (int i = 0; i < 8; ++i) tot += s_red[i];
    s_bcast[2] = tot * (1.0f / (float)DD);
  }
  __syncthreads();
  const float mu = s_bcast[2];

  const float q0 = e0 - mu, q1 = e1 - mu, q2 = e2 - mu;
  float v = q0 * q0 + q1 * q1 + q2 * q2;
  #pragma unroll
  for (int off = 16; off > 0; off >>= 1) v += __shfl_xor(v, off, 32);
  if (lane == 0) s_red[wv] = v;
  __syncthreads();
  if (tid == 0) {
    float tot = 0.f;
    #pragma unroll
    for (int i = 0; i < 8; ++i) tot += s_red[i];
    s_bcast[2] = tot * (1.0f / (float)DD);
  }
  __syncthreads();
  const float var  = s_bcast[2];
  const float rstd = rsqrtf(var + LN_EPS);
  const bool  has  = s_bcast[1] > 0.5f;

  float* __restrict__ obf = out + (size_t)bf * DD;
  {
    const int d = tid;
    obf[d] = has ? ((e0 - mu) * rstd * gamma[d] + beta[d]) : e0;
  }
  {
    const int d = tid + 256;
    obf[d] = has ? ((e1 - mu) * rstd * gamma[d] + beta[d]) : e1;
  }
  {
    const int d = tid + 512;
    obf[d] = has ? ((e2 - mu) * rstd * gamma[d] + beta[d]) : e2;
  }
}

extern "C" void kernel_launch(void* const* d_in, const int* in_sizes, int n_in,
                              void* d_out, int out_size, void* d_ws, size_t ws_size,
                              hipStream_t stream) {
  const float* H     = (const float*)d_in[0];
  const int*   vlens = (const int*)d_in[1];
  const float* gamma = (const float*)d_in[2];
  const float* beta  = (const float*)d_in[3];
  float* out = (float*)d_out;
  const int BF = in_sizes[1];           // B*F = 256
  mda_kernel<<<BF, 256, 0, stream>>>(H, vlens, gamma, beta, out);
}